// HyperbolicPrototypeHead_77910706749473
// MI455X (gfx1250) — compile-verified
//
#include <hip/hip_runtime.h>
#include <math.h>

// Problem constants from the reference.
#define BDIM 512
#define CDIM 512
#define DDIM 512

typedef __attribute__((ext_vector_type(2))) float v2f;
typedef __attribute__((ext_vector_type(8))) float v8f;

// c = clip(softplus(raw_c) + 1e-6, 1e-4, 1e4)
__device__ __forceinline__ float current_c(float raw) {
    float sp = (raw > 20.f) ? raw : log1pf(expf(raw));
    float c = sp + 1e-6f;
    return fminf(fmaxf(c, 1e-4f), 1e4f);
}

// ---------------------------------------------------------------------------
// Kernel 1: per-row expmap0 scale + squared norm for feats (B rows) and
// protos (C rows). One wave32 per row; butterfly shuffle reduction.
// ws layout (floats): [0,B)=scale_z  [B,2B)=x2  [2B,2B+C)=scale_p  [..+C)=y2
// ---------------------------------------------------------------------------
__global__ void __launch_bounds__(256)
row_stats_kernel(const float* __restrict__ feats,
                 const float* __restrict__ protos,
                 const float* __restrict__ raw_c,
                 float* __restrict__ ws) {
    int wave = (blockIdx.x * blockDim.x + threadIdx.x) >> 5;
    int lane = threadIdx.x & 31;
    if (wave >= BDIM + CDIM) return;

    const float* row = (wave < BDIM) ? (feats + (size_t)wave * DDIM)
                                     : (protos + (size_t)(wave - BDIM) * DDIM);
    float s = 0.f;
    #pragma unroll 4
    for (int j = lane; j < DDIM; j += 32) {
        float v = row[j];
        s = fmaf(v, v, s);
    }
    #pragma unroll
    for (int off = 16; off > 0; off >>= 1) s += __shfl_xor(s, off, 32);

    if (lane == 0) {
        float c  = current_c(raw_c[0]);
        float sc = sqrtf(c);
        float n2 = s;
        float n  = sqrtf(n2);
        float nc = fmaxf(n, 1e-15f);           // NORM_EPS clamp
        float scn = sc * nc;
        float scale = tanhf(scn) / scn;        // expmap0 row scale
        float sq = scale * scale * n2;         // ||z||^2 (== ||x||^2)
        if (wave < BDIM) {
            ws[wave]        = scale;
            ws[BDIM + wave] = sq;
        } else {
            int r = wave - BDIM;
            ws[2 * BDIM + r]        = scale;
            ws[2 * BDIM + CDIM + r] = sq;
        }
    }
}

// ---------------------------------------------------------------------------
// Kernel 2: fused fp32 WMMA GEMM (feats @ proto^T) + hyperbolic epilogue.
// Each wave computes a 32x32 output block as 2x2 tiles of
// V_WMMA_F32_16X16X4_F32 (exact fp32). 128-thread block -> 64x64 region.
//
// A fragment layout (16x4 f32): lane<16 holds A[M=lane, k..k+1],
// lane>=16 holds A[M=lane-16, k+2..k+3] -> float2 load at row*D + k + 2*half.
// B fragment (4x16, B[k][n] = proto[n][k]) uses the identical pattern on
// proto rows. C/D layout: vgpr r <-> row r + 8*half, col lane&15.
// ---------------------------------------------------------------------------
__global__ void __launch_bounds__(128)
gemm_head_kernel(const float* __restrict__ feats,
                 const float* __restrict__ protos,
                 const float* __restrict__ raw_c,
                 const float* __restrict__ ws,
                 float* __restrict__ out) {
    const int lane = threadIdx.x & 31;
    const int wave = threadIdx.x >> 5;                 // 0..3
    const int m0 = blockIdx.y * 64 + (wave >> 1) * 32; // batch rows
    const int n0 = blockIdx.x * 64 + (wave & 1) * 32;  // class cols

    const int half = lane >> 4;
    const int lm   = lane & 15;

    const float* aptr0 = feats  + (size_t)(m0 + lm) * DDIM + 2 * half;
    const float* aptr1 = aptr0 + 16 * DDIM;
    const float* bptr0 = protos + (size_t)(n0 + lm) * DDIM + 2 * half;
    const float* bptr1 = bptr0 + 16 * DDIM;

    v8f acc00 = {}, acc01 = {}, acc10 = {}, acc11 = {};

    #pragma unroll 4
    for (int k = 0; k < DDIM; k += 4) {
        v2f a0 = *(const v2f*)(aptr0 + k);
        v2f a1 = *(const v2f*)(aptr1 + k);
        v2f b0 = *(const v2f*)(bptr0 + k);
        v2f b1 = *(const v2f*)(bptr1 + k);
        acc00 = __builtin_amdgcn_wmma_f32_16x16x4_f32(false, a0, false, b0,
                                                      (short)0, acc00, false, false);
        acc01 = __builtin_amdgcn_wmma_f32_16x16x4_f32(false, a0, false, b1,
                                                      (short)0, acc01, false, false);
        acc10 = __builtin_amdgcn_wmma_f32_16x16x4_f32(false, a1, false, b0,
                                                      (short)0, acc10, false, false);
        acc11 = __builtin_amdgcn_wmma_f32_16x16x4_f32(false, a1, false, b1,
                                                      (short)0, acc11, false, false);
    }

    // ---- epilogue: mobius_add norm + artanh distance ----
    const float c   = current_c(raw_c[0]);
    const float sc  = sqrtf(c);
    const float inv = 2.f / sc;

    const float* scale_z = ws;
    const float* x2a     = ws + BDIM;
    const float* scale_p = ws + 2 * BDIM;
    const float* y2a     = ws + 2 * BDIM + CDIM;

    v8f accs[2][2] = { { acc00, acc01 }, { acc10, acc11 } };

    #pragma unroll
    for (int i = 0; i < 2; ++i) {
        #pragma unroll
        for (int j = 0; j < 2; ++j) {
            v8f acc = accs[i][j];
            const int col = n0 + 16 * j + lm;
            const float sp  = scale_p[col];
            const float y2v = y2a[col];
            #pragma unroll
            for (int r = 0; r < 8; ++r) {
                const int row = m0 + 16 * i + 8 * half + r;
                const float sz  = scale_z[row];
                const float x2v = x2a[row];
                const float dot = acc[r];              // u_b . v_c (fp32 exact)
                const float xy  = -(sz * sp) * dot;    // <x, p>, x = -z

                const float A = 1.f + 2.f * c * xy + c * y2v;
                const float E = 1.f - c * x2v;
                const float den = fmaxf(1.f + 2.f * c * xy + c * c * x2v * y2v,
                                        1e-15f);
                // ||num||^2 = A^2 x2 + 2AE<x,p> + E^2 y2
                float msq = fmaf(A * A, x2v, fmaf(2.f * A * E, xy, E * E * y2v));
                msq = fmaxf(msq, 0.f);
                const float mnorm = sqrtf(msq) / den;
                float t = sc * mnorm;
                t = fminf(fmaxf(t, -1.f + 1e-7f), 1.f - 1e-7f);
                const float d = inv * atanhf(t);
                out[(size_t)row * CDIM + col] = -d;    // TAU == 1
            }
        }
    }
}

// ---------------------------------------------------------------------------
extern "C" void kernel_launch(void* const* d_in, const int* in_sizes, int n_in,
                              void* d_out, int out_size, void* d_ws, size_t ws_size,
                              hipStream_t stream) {
    const float* feats  = (const float*)d_in[0]; // (B, D)
    const float* protos = (const float*)d_in[1]; // (C, D)
    const float* raw_c  = (const float*)d_in[2]; // scalar
    float* out = (float*)d_out;                  // (B, C)
    float* ws  = (float*)d_ws;                   // 2*(B+C) floats

    // Phase 1: row norms / expmap0 scales (one wave per row).
    const int total_rows = BDIM + CDIM;
    const int threads1 = 256;                    // 8 waves/block
    const int blocks1 = (total_rows + (threads1 / 32) - 1) / (threads1 / 32);
    row_stats_kernel<<<blocks1, threads1, 0, stream>>>(feats, protos, raw_c, ws);

    // Phase 2: fused WMMA GEMM + epilogue. 64x64 per 128-thread block.
    dim3 grid(CDIM / 64, BDIM / 64);
    gemm_head_kernel<<<grid, 128, 0, stream>>>(feats, protos, raw_c, ws, out);
}